// CheckpointedNoseHoover_12824772346266
// MI455X (gfx1250) — compile-verified
//
#include <hip/hip_runtime.h>

#define NH_DT 0.001f
#define LOG2E 1.4426950408889634f

// Wave32 full butterfly reduction via ds_swizzle xor patterns (group-of-32
// mode: offset[14:10]=xor_mask, offset[4:0]=and_mask=0x1F). No bounds-check
// VALU overhead. EXEC is all-ones for active waves, so swizzle reads are valid.
__device__ __forceinline__ float wave_sum32(float x) {
  x += __int_as_float(__builtin_amdgcn_ds_swizzle(__float_as_int(x), 0x041F)); // xor 1
  x += __int_as_float(__builtin_amdgcn_ds_swizzle(__float_as_int(x), 0x081F)); // xor 2
  x += __int_as_float(__builtin_amdgcn_ds_swizzle(__float_as_int(x), 0x101F)); // xor 4
  x += __int_as_float(__builtin_amdgcn_ds_swizzle(__float_as_int(x), 0x201F)); // xor 8
  x += __int_as_float(__builtin_amdgcn_ds_swizzle(__float_as_int(x), 0x401F)); // xor 16
  return x;
}

__launch_bounds__(256)
__global__ void nose_hoover_traj_kernel(const float* __restrict__ x0g,
                                        const float* __restrict__ v0g,
                                        const float* __restrict__ alpha0,
                                        const float* __restrict__ kTp,
                                        const float* __restrict__ massp,
                                        const float* __restrict__ Qp,
                                        const int*  __restrict__ nstepsp,
                                        const int*  __restrict__ storep,
                                        float* __restrict__ out,
                                        int B) {
  constexpr int D = 64;                 // DOF per system (2 per lane, wave32)
  // Double-buffered per-wave staging for async LDS->global snapshot stores.
  // [buf][wave-in-block][x=0/v=1][lane] : 2*8*2*32*8B = 8 KB
  __shared__ float2 stg[2][8][2][32];

  const int lane = threadIdx.x & 31;
  const int wv   = threadIdx.x >> 5;
  const int sys  = blockIdx.x * 8 + wv;   // one wave per system (uniform per wave)
  if (sys >= B) return;

  const float kT          = kTp[0];
  const float mass        = massp[0];
  const float Q           = Qp[0];
  const int   n_steps     = nstepsp[0];
  const int   store_every = storep[0];
  const int   n_chunks    = n_steps / store_every;

  const float c      = (0.25f * NH_DT) / Q;       // DT/4 / Q
  const float dconst = (float)D * kT;             // ndof * kT
  const float hdtm   = (0.5f * NH_DT) / mass;     // DT/2 / mass  (force = -x)
  const float k2     = -(0.5f * NH_DT) * LOG2E;   // exp(-a*DT/2) == exp2(a*k2)

  const size_t base = (size_t)sys * D + (size_t)(lane * 2);
  float2 xp = *(const float2*)(x0g + base);
  float2 vp = *(const float2*)(v0g + base);
  float x0 = xp.x, x1 = xp.y;
  float v0 = vp.x, v1 = vp.y;
  float alpha = alpha0[sys];

  float* const outx = out;
  float* const outv = out + (size_t)(n_chunks + 1) * (size_t)B * D;

  // Snapshot via CDNA5 async store path: stage to LDS, fire
  // global_store_async_from_lds_b64 (tracked by ASYNCcnt), keep integrating.
  auto snapshot = [&](int chunk) {
    const int buf = chunk & 1;
    // Before reusing this buffer, drain stores issued 2 snapshots ago
    // (<=2 outstanding == only the other buffer's pair may remain in flight).
    asm volatile("s_wait_asynccnt 2" ::: "memory");
    stg[buf][wv][0][lane] = make_float2(x0, x1);
    stg[buf][wv][1][lane] = make_float2(v0, v1);
    // LDS writes must land before the async engine reads them.
    asm volatile("s_wait_dscnt 0" ::: "memory");
    const size_t off = ((size_t)chunk * (size_t)B + (size_t)sys) * D + (size_t)(lane * 2);
    unsigned long long gax = (unsigned long long)(outx + off);
    unsigned long long gav = (unsigned long long)(outv + off);
    unsigned lx = (unsigned)(size_t)&stg[buf][wv][0][lane];  // low 32 bits = LDS offset
    unsigned lv = (unsigned)(size_t)&stg[buf][wv][1][lane];
    asm volatile("global_store_async_from_lds_b64 %0, %1, off"
                 :: "v"(gax), "v"(lx) : "memory");
    asm volatile("global_store_async_from_lds_b64 %0, %1, off"
                 :: "v"(gav), "v"(lv) : "memory");
  };

  snapshot(0);  // traj[0] = (x0, v0)

  // Carry Sigma v^2 across steps (v is unchanged between the last and first
  // alpha-updates of consecutive steps -> bitwise-identical reuse).
  float v2 = wave_sum32(fmaf(v0, v0, v1 * v1));

  for (int chunk = 1; chunk <= n_chunks; ++chunk) {
    for (int s = store_every; s > 0; --s) {
      // alpha quarter-kick 1 (reuses carried v2)
      alpha = fmaf(c, v2 - dconst, alpha);
      // thermostat scale 1; Sigma(v*sc)^2 == sc^2 * Sigma v^2 (uniform scale)
      float sc = __builtin_amdgcn_exp2f(alpha * k2);   // raw v_exp_f32
      v0 *= sc; v1 *= sc;
      v2 *= sc * sc;
      alpha = fmaf(c, v2 - dconst, alpha);
      // velocity-Verlet kick-drift-kick with F(x) = -x
      v0 = fmaf(-hdtm, x0, v0);
      v1 = fmaf(-hdtm, x1, v1);
      x0 = fmaf(NH_DT, v0, x0);
      x1 = fmaf(NH_DT, v1, x1);
      v0 = fmaf(-hdtm, x0, v0);
      v1 = fmaf(-hdtm, x1, v1);
      // the one real cross-lane reduction per step
      v2 = wave_sum32(fmaf(v0, v0, v1 * v1));
      alpha = fmaf(c, v2 - dconst, alpha);
      // thermostat scale 2
      sc = __builtin_amdgcn_exp2f(alpha * k2);
      v0 *= sc; v1 *= sc;
      v2 *= sc * sc;
      alpha = fmaf(c, v2 - dconst, alpha);
    }
    snapshot(chunk);
  }
  // Drain async stores (S_ENDPGM also implicitly waits, this is belt+braces).
  asm volatile("s_wait_asynccnt 0" ::: "memory");
}

extern "C" void kernel_launch(void* const* d_in, const int* in_sizes, int n_in,
                              void* d_out, int out_size, void* d_ws, size_t ws_size,
                              hipStream_t stream) {
  (void)n_in; (void)out_size; (void)d_ws; (void)ws_size;
  const float* x0   = (const float*)d_in[0];
  const float* v0   = (const float*)d_in[1];
  const float* a0   = (const float*)d_in[2];
  const float* kT   = (const float*)d_in[3];
  const float* mass = (const float*)d_in[4];
  const float* Q    = (const float*)d_in[5];
  const int*   ns   = (const int*)d_in[6];
  const int*   se   = (const int*)d_in[7];

  const int D = 64;
  const int B = in_sizes[0] / D;        // 4096 systems
  dim3 block(256);                      // 8 wave32 waves = 8 systems per block
  dim3 grid((unsigned)((B + 7) / 8));

  nose_hoover_traj_kernel<<<grid, block, 0, stream>>>(
      x0, v0, a0, kT, mass, Q, ns, se, (float*)d_out, B);
}